// NTXentLoss_6227702579661
// MI455X (gfx1250) — compile-verified
//
#include <hip/hip_runtime.h>

typedef __attribute__((ext_vector_type(2))) float v2f;
typedef __attribute__((ext_vector_type(8))) float v8f;

#define B_SZ   4096
#define TWOB   8192
#define D_DIM  128
#define INV_T  10.0f            // 1 / temperature
#define RB     128              // rows per block (4 waves x 32 rows)
#define CS     1024             // columns per slice
#define NS     (TWOB / CS)      // 8 column slices
#define NTILE  (CS / 16)        // 64 column tiles per slice
#define LDSW   132              // padded LDS tile row stride (words) -> conflict-free

// ---------------------------------------------------------------------------
// Kernel 1: row-normalize reps = [zjs ; zis] into workspace (fp32).
// ---------------------------------------------------------------------------
__global__ __launch_bounds__(256) void nrm_kernel(const float* __restrict__ zis,
                                                  const float* __restrict__ zjs,
                                                  float* __restrict__ nrm) {
    const int wave = threadIdx.x >> 5;
    const int lane = threadIdx.x & 31;
    const int row  = blockIdx.x * 8 + wave;
    const float* src = (row < B_SZ) ? (zjs + (size_t)row * D_DIM)
                                    : (zis + (size_t)(row - B_SZ) * D_DIM);
    float4 v = ((const float4*)src)[lane];
    float ss = v.x * v.x + v.y * v.y + v.z * v.z + v.w * v.w;
    #pragma unroll
    for (int off = 16; off > 0; off >>= 1) ss += __shfl_xor(ss, off, 32);
    const float scale = 1.0f / fmaxf(sqrtf(ss), 1e-8f);
    float4 o; o.x = v.x * scale; o.y = v.y * scale; o.z = v.z * scale; o.w = v.w * scale;
    ((float4*)(nrm + (size_t)row * D_DIM))[lane] = o;
}

// ---------------------------------------------------------------------------
// Kernel 2: fused sim + fixed-max sum-exp over a column slice.
// Block = 128 threads (4 waves). Block owns rows [m0, m0+128) and columns
// [cbase, cbase+1024). Wave owns 2 row tiles (A fragments in registers).
// B tiles staged global->LDS, double-buffered; 2 WMMAs per ds_load_b64.
// ---------------------------------------------------------------------------
__global__ __launch_bounds__(128) void ntxent_kernel(const float* __restrict__ nrm,
                                                     float* __restrict__ rowsum,
                                                     float* __restrict__ pospart) {
    __shared__ float tile[2][16][LDSW];      // double-buffered 16x128 B tile
    __shared__ float sm[4][2][16][16];       // per-wave row partials
    __shared__ float spos[4];

    const int tid    = threadIdx.x;
    const int wave   = tid >> 5;
    const int lane   = tid & 31;
    const int lane16 = lane & 15;
    const int half   = lane >> 4;
    const int m0     = blockIdx.x * RB;
    const int cbase  = blockIdx.y * CS;

    // --- A fragments for two 16-row tiles (kept in VGPRs for all 64 tiles) ---
    const int ar0 = m0 + wave * 32 + lane16;
    const int ar1 = ar0 + 16;
    v2f a0[32], a1[32];
    #pragma unroll
    for (int kk = 0; kk < 32; ++kk) {
        a0[kk] = *(const v2f*)(nrm + (size_t)ar0 * D_DIM + kk * 4 + half * 2);
        a1[kk] = *(const v2f*)(nrm + (size_t)ar1 * D_DIM + kk * 4 + half * 2);
    }

    // staging role: thread -> (row-of-tile, 16-float segment)
    const int srow = tid >> 3;   // 0..15
    const int sseg = tid & 7;    // 0..7

    // prologue: stage tile 0 into buffer 0
    {
        const float* g = nrm + (size_t)(cbase + srow) * D_DIM + sseg * 16;
        #pragma unroll
        for (int q = 0; q < 4; ++q)
            *(float4*)&tile[0][srow][sseg * 16 + q * 4] = ((const float4*)g)[q];
    }
    __syncthreads();

    v8f acc_s0 = {}, acc_s1 = {};
    float possum = 0.0f;

    for (int it = 0; it < NTILE; ++it) {
        const int  p    = it & 1;
        const bool more = (it + 1) < NTILE;

        // issue next tile's global loads early; WMMA chain hides their latency
        float4 st[4];
        if (more) {
            const float* g = nrm + (size_t)(cbase + (it + 1) * 16 + srow) * D_DIM + sseg * 16;
            #pragma unroll
            for (int q = 0; q < 4; ++q) st[q] = ((const float4*)g)[q];
        }
        if (it + 2 < NTILE)
            __builtin_prefetch(nrm + (size_t)(cbase + (it + 2) * 16 + srow) * D_DIM, 0, 1);

        // 16x16 sim tiles: K=128 in 32 steps, B fragment shared by 2 WMMAs
        v8f c0 = {}, c1 = {};
        #pragma unroll
        for (int kk = 0; kk < 32; ++kk) {
            v2f b = *(const v2f*)&tile[p][lane16][kk * 4 + half * 2];
            c0 = __builtin_amdgcn_wmma_f32_16x16x4_f32(false, a0[kk], false, b, (short)0, c0, false, false);
            c1 = __builtin_amdgcn_wmma_f32_16x16x4_f32(false, a1[kk], false, b, (short)0, c1, false, false);
        }

        // fixed-max softmax accumulation (max = 10 since |cos|<=1)
        const int colg = cbase + it * 16 + lane16;
        #pragma unroll
        for (int v = 0; v < 8; ++v) {
            const int r0 = m0 + wave * 32 + v + (half << 3);   // C layout: M = v + 8*half
            const int r1 = r0 + 16;
            const float s0 = c0[v], s1 = c1[v];
            const float e0 = (colg == r0) ? 0.0f : __expf(INV_T * s0 - INV_T);
            const float e1 = (colg == r1) ? 0.0f : __expf(INV_T * s1 - INV_T);
            acc_s0[v] += e0;
            acc_s1[v] += e1;
            if (colg == ((r0 + B_SZ) & (TWOB - 1))) possum += INV_T * s0;
            if (colg == ((r1 + B_SZ) & (TWOB - 1))) possum += INV_T * s1;
        }

        // drain staged registers into the back buffer
        if (more) {
            #pragma unroll
            for (int q = 0; q < 4; ++q)
                *(float4*)&tile[p ^ 1][srow][sseg * 16 + q * 4] = st[q];
        }
        __syncthreads();
    }

    // ---- block reduction: per-row sums + positive-logit partial ----
    #pragma unroll
    for (int v = 0; v < 8; ++v) {
        sm[wave][0][v + (half << 3)][lane16] = acc_s0[v];
        sm[wave][1][v + (half << 3)][lane16] = acc_s1[v];
    }
    #pragma unroll
    for (int off = 16; off > 0; off >>= 1) possum += __shfl_xor(possum, off, 32);
    if (lane == 0) spos[wave] = possum;
    __syncthreads();

    {   // 128 threads -> one local row each
        const int row = tid;                  // 0..127
        const int w = row >> 5, tt = (row >> 4) & 1, r = row & 15;
        float S = 0.0f;
        #pragma unroll
        for (int cidx = 0; cidx < 16; ++cidx) S += sm[w][tt][r][cidx];
        rowsum[(size_t)blockIdx.y * TWOB + m0 + row] = S;
    }
    if (tid == 0)
        pospart[blockIdx.y * gridDim.x + blockIdx.x] = spos[0] + spos[1] + spos[2] + spos[3];
}

// ---------------------------------------------------------------------------
// Kernel 3: deterministic finalization: loss = (sum_r [log S_r + 10] - sum pos)/2B
// ---------------------------------------------------------------------------
__global__ __launch_bounds__(256) void finish_kernel(const float* __restrict__ rowsum,
                                                     const float* __restrict__ pospart,
                                                     float* __restrict__ out) {
    __shared__ float red[256];
    float acc = 0.0f;
    for (int r = threadIdx.x; r < TWOB; r += 256) {
        float S = 0.0f;
        #pragma unroll
        for (int s = 0; s < NS; ++s) S += rowsum[(size_t)s * TWOB + r];
        acc += __logf(S) + INV_T;
    }
    for (int i = threadIdx.x; i < NS * (TWOB / RB); i += 256) acc -= pospart[i];
    red[threadIdx.x] = acc;
    __syncthreads();
    for (int off = 128; off > 0; off >>= 1) {
        if (threadIdx.x < off) red[threadIdx.x] += red[threadIdx.x + off];
        __syncthreads();
    }
    if (threadIdx.x == 0) out[0] = red[0] / (float)TWOB;
}

extern "C" void kernel_launch(void* const* d_in, const int* in_sizes, int n_in,
                              void* d_out, int out_size, void* d_ws, size_t ws_size,
                              hipStream_t stream) {
    const float* zis = (const float*)d_in[0];
    const float* zjs = (const float*)d_in[1];

    float* nrm      = (float*)d_ws;                      // 8192*128 f32 = 4 MB
    float* rowsum   = nrm + (size_t)TWOB * D_DIM;        // 8*8192 f32 = 256 KB
    float* pospart  = rowsum + (size_t)NS * TWOB;        // 512 f32

    nrm_kernel<<<TWOB / 8, 256, 0, stream>>>(zis, zjs, nrm);

    dim3 grid2(TWOB / RB, NS);                           // (64, 8)
    ntxent_kernel<<<grid2, 128, 0, stream>>>(nrm, rowsum, pospart);

    finish_kernel<<<1, 256, 0, stream>>>(rowsum, pospart, (float*)d_out);
}